// SparseFusionThorMoE_7533372638014
// MI455X (gfx1250) — compile-verified
//
#include <hip/hip_runtime.h>
#include <math.h>

// Problem dims (fixed by reference)
#define E_  16
#define T_  8192
#define H_  1024
#define I_  4096
#define C_  (T_ / E_)      // 512 tokens per expert

// GEMM tiling
#define BM   128
#define BN   128
#define BK   32
#define LDA_ 36            // padded LDS stride for A (float4-aligned, conflict-free)
#define LDB_ 136           // padded LDS stride for B (float4-aligned, conflict-free)

typedef __attribute__((ext_vector_type(2))) float v2f;
typedef __attribute__((ext_vector_type(4))) float f4v;
typedef __attribute__((ext_vector_type(8))) float v8f;

// ---- CDNA5 async global->LDS path (GLOBAL_LOAD_ASYNC_TO_LDS_B128, ASYNCcnt) ----
#if __has_builtin(__builtin_amdgcn_global_load_async_to_lds_b128)
#define HAS_ASYNC_LDS 1
#else
#define HAS_ASYNC_LDS 0
#endif

#if HAS_ASYNC_LDS
typedef int v4i_ __attribute__((vector_size(16)));
typedef __attribute__((address_space(1))) v4i_* gptr_v4i;   // global (__device__)
typedef __attribute__((address_space(3))) v4i_* lptr_v4i;   // LDS (__shared__)

__device__ __forceinline__ void async_cp16(const float* g, float* l)
{
    __builtin_amdgcn_global_load_async_to_lds_b128(
        (gptr_v4i)(g), (lptr_v4i)(l), 0, 0);
}
__device__ __forceinline__ void async_wait0()
{
#if __has_builtin(__builtin_amdgcn_s_wait_asynccnt)
    __builtin_amdgcn_s_wait_asynccnt(0);
#else
    asm volatile("s_wait_asynccnt 0" ::: "memory");
#endif
}
#endif

// D = A(row-major, gathered rows) x B(row-major) + bias, optional exact GELU,
// optional scatter of output rows through perm.
// A logical shape per expert: [C_, Kdim]; B per expert: [Kdim, Ndim].
template <int Kdim, int Ndim, bool GATHER_A, bool DO_GELU, bool SCATTER_OUT>
__global__ __launch_bounds__(256)
void moe_gemm(const float* __restrict__ A,
              const float* __restrict__ Bw,
              const float* __restrict__ bias,
              const long long* __restrict__ perm,
              float* __restrict__ Out)
{
    __shared__ float As[2][BM][LDA_];
    __shared__ float Bs[2][BK][LDB_];
    __shared__ int   rowMap[BM];    // global A row per tile row
    __shared__ int   scatMap[BM];   // global Out row per tile row (scatter)

    const int e    = blockIdx.z;
    const int m0   = blockIdx.y * BM;   // row offset within expert [0, C_)
    const int n0   = blockIdx.x * BN;
    const int tid  = threadIdx.x;
    const int lane = tid & 31;
    const int wave = tid >> 5;
    const int wm   = wave >> 2;         // 0..1 : wave row   (64 rows each)
    const int wn   = wave & 3;          // 0..3 : wave col   (32 cols each)
    const int mlan = lane & 15;
    const int hi   = lane >> 4;         // half-wave select

    if (tid < BM) {
        const int r = e * C_ + m0 + tid;
        rowMap[tid]  = GATHER_A    ? (int)perm[r] : r;
        scatMap[tid] = SCATTER_OUT ? (int)perm[r] : r;
    }
    __syncthreads();

    const float* __restrict__ Be = Bw + (size_t)e * Kdim * Ndim;

    // Per-thread staging registers (used only by the non-async fallback)
    f4v regA[4], regB[4];
    (void)regA; (void)regB;

    // issue: start the global->LDS movement for tile at k0 into buffer `buf`
    auto issue_tile = [&](int k0, int buf) {
#pragma unroll
        for (int i = 0; i < (BM * BK) / (256 * 4); ++i) {     // A: 4 per thread
            const int v   = tid + i * 256;
            const int row = v >> 3;                           // BK/4 = 8 vec4/row
            const int kc  = (v & 7) << 2;
            const float* g = A + (size_t)rowMap[row] * Kdim + k0 + kc;
#if HAS_ASYNC_LDS
            async_cp16(g, &As[buf][row][kc]);
#else
            regA[i] = *(const f4v*)g;
#endif
        }
#pragma unroll
        for (int i = 0; i < (BK * BN) / (256 * 4); ++i) {     // B: 4 per thread
            const int v  = tid + i * 256;
            const int kr = v >> 5;                            // BN/4 = 32 vec4/row
            const int nc = (v & 31) << 2;
            const float* g = Be + (size_t)(k0 + kr) * Ndim + n0 + nc;
#if HAS_ASYNC_LDS
            async_cp16(g, &Bs[buf][kr][nc]);
#else
            regB[i] = *(const f4v*)g;
#endif
        }
    };
    // commit: (fallback only) register -> LDS stores after compute overlap
    auto commit_tile = [&](int buf) {
        (void)buf;
#if !HAS_ASYNC_LDS
#pragma unroll
        for (int i = 0; i < (BM * BK) / (256 * 4); ++i) {
            const int v = tid + i * 256;
            *(f4v*)&As[buf][v >> 3][(v & 7) << 2] = regA[i];
        }
#pragma unroll
        for (int i = 0; i < (BK * BN) / (256 * 4); ++i) {
            const int v = tid + i * 256;
            *(f4v*)&Bs[buf][v >> 5][(v & 31) << 2] = regB[i];
        }
#endif
    };
    auto wait_tile = [&]() {
#if HAS_ASYNC_LDS
        async_wait0();
#endif
    };

    v8f acc[4][2];
    const v8f vzero = {0.f, 0.f, 0.f, 0.f, 0.f, 0.f, 0.f, 0.f};
#pragma unroll
    for (int mt = 0; mt < 4; ++mt)
#pragma unroll
        for (int nt = 0; nt < 2; ++nt)
            acc[mt][nt] = vzero;

    // ---- software pipeline: prologue loads tile 0 ----
    issue_tile(0, 0);
    commit_tile(0);
    wait_tile();
    __syncthreads();

    int cur = 0;
    for (int k0 = 0; k0 < Kdim; k0 += BK) {
        const bool has_next = (k0 + BK) < Kdim;
        if (has_next) issue_tile(k0 + BK, cur ^ 1);           // overlap with WMMA

        // ---- 8 k-steps of V_WMMA_F32_16X16X4_F32 on buffer `cur` ----
#pragma unroll
        for (int kk = 0; kk < BK; kk += 4) {
            const int kh = kk + (hi << 1);     // lane<16 -> K{0,1}; lane>=16 -> K{2,3}
            v2f a[4], b[2];
#pragma unroll
            for (int mt = 0; mt < 4; ++mt) {
                const int row = wm * 64 + mt * 16 + mlan;
                a[mt].x = As[cur][row][kh];
                a[mt].y = As[cur][row][kh + 1];
            }
#pragma unroll
            for (int nt = 0; nt < 2; ++nt) {
                const int col = wn * 32 + nt * 16 + mlan;
                b[nt].x = Bs[cur][kh][col];
                b[nt].y = Bs[cur][kh + 1][col];
            }
#pragma unroll
            for (int mt = 0; mt < 4; ++mt)
#pragma unroll
                for (int nt = 0; nt < 2; ++nt)
                    acc[mt][nt] = __builtin_amdgcn_wmma_f32_16x16x4_f32(
                        false, a[mt], false, b[nt], (short)0, acc[mt][nt],
                        false, false);
        }

        if (has_next) commit_tile(cur ^ 1);
        wait_tile();                                          // our async copies landed
        __syncthreads();                                      // everyone's landed
        cur ^= 1;
    }

    // ---- epilogue: bias (+GELU), (scattered) store ----
    const float* __restrict__ be = bias + (size_t)e * Ndim;
    float bn[2];
#pragma unroll
    for (int nt = 0; nt < 2; ++nt)
        bn[nt] = be[n0 + wn * 32 + nt * 16 + mlan];

#pragma unroll
    for (int mt = 0; mt < 4; ++mt) {
#pragma unroll
        for (int nt = 0; nt < 2; ++nt) {
            const int n = n0 + wn * 32 + nt * 16 + mlan;
#pragma unroll
            for (int i = 0; i < 8; ++i) {
                // C/D layout: VGPR i, lanes 0-15 -> M=i, lanes 16-31 -> M=i+8
                const int mloc = wm * 64 + mt * 16 + i + (hi << 3);
                float v = acc[mt][nt][i] + bn[nt];
                if (DO_GELU)
                    v = 0.5f * v * (1.0f + erff(v * 0.70710678118654752f));
                const size_t orow = (size_t)scatMap[mloc];
                Out[orow * Ndim + n] = v;
            }
        }
    }
}

// Residual add + LayerNorm over H_=1024, one 256-thread block per token.
__global__ __launch_bounds__(256)
void ln_residual(const float* __restrict__ y, const float* __restrict__ resid,
                 const float* __restrict__ w, const float* __restrict__ b,
                 float* __restrict__ out)
{
    const int t = blockIdx.x;
    const float* __restrict__ yr = y + (size_t)t * H_;
    const float* __restrict__ hr = resid + (size_t)t * H_;

    float vals[H_ / 256];
    float s = 0.f, ss = 0.f;
#pragma unroll
    for (int i = 0; i < H_ / 256; ++i) {
        const int n = threadIdx.x + i * 256;
        const float v = yr[n] + hr[n];
        vals[i] = v;
        s  += v;
        ss += v * v;
    }
    // wave32 reduce
#pragma unroll
    for (int off = 16; off > 0; off >>= 1) {
        s  += __shfl_down(s, off, 32);
        ss += __shfl_down(ss, off, 32);
    }
    __shared__ float red[16];
    const int wave = threadIdx.x >> 5;
    if ((threadIdx.x & 31) == 0) { red[wave] = s; red[wave + 8] = ss; }
    __syncthreads();
    float S = 0.f, SS = 0.f;
#pragma unroll
    for (int wv = 0; wv < 8; ++wv) { S += red[wv]; SS += red[wv + 8]; }

    const float mu  = S * (1.0f / H_);
    const float var = SS * (1.0f / H_) - mu * mu;
    const float inv = rsqrtf(var + 1e-12f);
#pragma unroll
    for (int i = 0; i < H_ / 256; ++i) {
        const int n = threadIdx.x + i * 256;
        out[(size_t)t * H_ + n] = (vals[i] - mu) * inv * w[n] + b[n];
    }
}

extern "C" void kernel_launch(void* const* d_in, const int* in_sizes, int n_in,
                              void* d_out, int out_size, void* d_ws, size_t ws_size,
                              hipStream_t stream)
{
    const float*     hidden = (const float*)d_in[0];
    const float*     w1     = (const float*)d_in[1];
    const float*     b1     = (const float*)d_in[2];
    const float*     w2     = (const float*)d_in[3];
    const float*     b2     = (const float*)d_in[4];
    const float*     lnw    = (const float*)d_in[5];
    const float*     lnb    = (const float*)d_in[6];
    const long long* perm   = (const long long*)d_in[7];

    float* act = (float*)d_ws;                 // [T_, I_]  (shuffled order) 128 MB
    float* y   = act + (size_t)T_ * I_;        // [T_, H_]  (restored order)  32 MB

    const dim3 blk(256);
    // GEMM1: gather + x@W1 + b1 + GELU -> act
    moe_gemm<H_, I_, true, true, false>
        <<<dim3(I_ / BN, C_ / BM, E_), blk, 0, stream>>>(hidden, w1, b1, perm, act);
    // GEMM2: act@W2 + b2, scatter-restore -> y
    moe_gemm<I_, H_, false, false, true>
        <<<dim3(H_ / BN, C_ / BM, E_), blk, 0, stream>>>(act, w2, b2, perm, y);
    // LayerNorm(y + hidden) -> out
    ln_residual<<<dim3(T_), blk, 0, stream>>>(y, hidden, lnw, lnb, (float*)d_out);
}